// TemporalLayer_57071525429388
// MI455X (gfx1250) — compile-verified
//
#include <hip/hip_runtime.h>
#include <math.h>

// AlwaysLayer forward for MI455X (gfx1250, wave32).
//   B=4096, K=8, bin1=8, bin_t=16, n=16.
// Per (b,k,j) group: 16 outputs, each a 16-bin reduction of log(1-(1-x)W).
// Transcendental-bound (67M v_log_f32); memory is only ~33 MB (~1.4us at
// 23.3 TB/s). Time-axis summation runs in f32 on the matrix pipe via
// V_WMMA_F32_16X16X4_F32 with an all-ones A matrix (column sums of B),
// which leaves the result lane-indexed by the node id n = lane%16.
// Inner-loop algebra: X = 1-(1-x)W = (1-W) + W*x, with W and (1-W) held in
// registers -> each element is exactly one v_fma_f32 + one v_log_f32.

typedef __attribute__((ext_vector_type(2))) float v2f;
typedef __attribute__((ext_vector_type(8))) float v8f;

#define K_DIM        8
#define BIN1         8
#define BIN_T        16
#define N_DIM        16
#define B_TOT        4096
#define B_CHUNKS     32                        // batch chunks per (k,j) pair
#define B_PER_BLOCK  (B_TOT / B_CHUNKS)        // 128
#define WAVES        8
#define ITERS        (B_PER_BLOCK / WAVES)     // 16

__global__ __launch_bounds__(256, 4)
void always_forward(const float* __restrict__ x,
                    const float* __restrict__ w_start,
                    const float* __restrict__ w_end,
                    float* __restrict__ out) {
  const int pair  = blockIdx.x >> 5;   // k*8 + j, 0..63
  const int chunk = blockIdx.x & 31;   // batch chunk, 0..31
  const int kk = pair >> 3;
  const int jj = pair & 7;
  const int tid  = threadIdx.x;
  const int wave = tid >> 5;
  const int lane = tid & 31;
  const int n = lane & 15;             // output node id
  const int h = lane >> 4;             // wave half -> which t sub-pair

  // W tile for this (k,j): [t][n], 256 floats. Computed once per block
  // (one v_exp_f32 per thread; 0.5M sigmoids total vs 67M logs -> free).
  __shared__ float Wlds[BIN_T * N_DIM];
  {
    const int t  = tid >> 4;
    const int nn = tid & 15;
    const float wsv = w_start[pair * 16 + nn];
    const float wev = w_end[pair * 16 + nn];
    const float z = ((float)t - wsv) * (wev - (float)t);
    Wlds[tid] = __builtin_amdgcn_rcpf(1.0f + __expf(-z));  // sigmoid(z)
  }
  __syncthreads();

  // This lane's 8 weights (chunks c=0..3, e=0..1, t = 4c + 2h + e) plus
  // their complements, so the hot loop is pure fma+log.
  float wreg[8], omw[8];
#pragma unroll
  for (int c = 0; c < 4; ++c) {
#pragma unroll
    for (int e = 0; e < 2; ++e) {
      const int t = 4 * c + 2 * h + e;
      const float w = Wlds[t * 16 + n];
      wreg[c * 2 + e] = w;
      omw[c * 2 + e]  = 1.0f - w;
    }
  }

  const v2f ones = {1.0f, 1.0f};       // A matrix (16x4 of ones)
  const size_t outCol = (size_t)pair * 16 + n;

#pragma unroll 2
  for (int it = 0; it < ITERS; ++it) {
    const int b = chunk * B_PER_BLOCK + it * WAVES + wave;
    // x4[b,k,j,t] = x[b, k*128 + t*8 + j]; fold h*16 into base so the two
    // loads per chunk use compile-time immediate offsets.
    const float* xl = x + (size_t)b * 1024 + kk * 128 + jj + h * 16;

    v8f acc = {0.f, 0.f, 0.f, 0.f, 0.f, 0.f, 0.f, 0.f};
#pragma unroll
    for (int c = 0; c < 4; ++c) {
      const float x0 = xl[c * 32 + 0];   // t = 4c + 2h
      const float x1 = xl[c * 32 + 8];   // t = 4c + 2h + 1
      // X = (1-W) + W*x  ==  1 - (1-x)*W
      const float l0 = __log2f(fmaf(x0, wreg[c * 2 + 0], omw[c * 2 + 0]));
      const float l1 = __log2f(fmaf(x1, wreg[c * 2 + 1], omw[c * 2 + 1]));
      v2f bv;
      bv.x = l0;
      bv.y = l1;
      // D = ones(16x4) x logX(4x16) + C  ->  D[m,n] = sum_t log2 X[t,n]
      acc = __builtin_amdgcn_wmma_f32_16x16x4_f32(
          false, ones, false, bv, (short)0, acc, false, false);
    }
    const float s = acc[0];            // column sum for this lane's n
    // y = 1/(1 - ln2 * s): single v_fma + v_rcp_f32 on the trans pipe.
    const float y = __builtin_amdgcn_rcpf(fmaf(-0.6931471805599453f, s, 1.0f));
    if (lane < 16) {                   // both halves hold identical results
      out[(size_t)b * 1024 + outCol] = y;
    }
  }
}

extern "C" void kernel_launch(void* const* d_in, const int* in_sizes, int n_in,
                              void* d_out, int out_size, void* d_ws, size_t ws_size,
                              hipStream_t stream) {
  (void)in_sizes; (void)n_in; (void)d_ws; (void)ws_size; (void)out_size;
  const float* x  = (const float*)d_in[0];
  const float* ws = (const float*)d_in[1];
  const float* we = (const float*)d_in[2];
  float* out = (float*)d_out;

  dim3 grid(64 * B_CHUNKS);   // 2048 blocks
  dim3 block(256);            // 8 waves (wave32)
  always_forward<<<grid, block, 0, stream>>>(x, ws, we, out);
}